// simpleEdgePrediction_1176821039616
// MI455X (gfx1250) — compile-verified
//
#include <hip/hip_runtime.h>
#include <math.h>

#define N_NODES_C  100000
#define N_EDGES_C  1000000
#define D_NODE_C   64
#define D_EDGE_C   16
#define NUM_PAR_C  80      // D_NODE + D_EDGE
#define HIDDEN_C   160     // 2 * NUM_PAR

typedef __attribute__((ext_vector_type(2))) float v2f;
typedef __attribute__((ext_vector_type(8))) float v8f;

// ---------------------------------------------------------------------------
// Kernel 1: collapse the linear MLP (no inter-layer nonlinearity, dropout is
// identity in eval), so:
//   w_eff[0..79] = W1 (80x160, row-major) @ W2 (160x1)
//   ws[80]       = b_eff = dot(b1, W2) + b2
// Done with V_WMMA_F32_16X16X4_F32 on a single wave32.
// ISA 32-bit layouts (CDNA5 ISA §7.12.2):
//   A (16x4, MxK): lanes 0-15 -> M=lane,   v0=K0, v1=K1
//                  lanes16-31 -> M=lane-16,v0=K2, v1=K3
//   B (4x16, KxN): element (k,n): lane = n + 16*(k>=2), vgpr = k&1
//                  -> only column N=0 is nonzero here (lane 0 / lane 16)
//   D (16x16):     element (m,0): lane = 16*(m>=8), vgpr = m&7
// All loads are unconditional; the B-column masking is a VALU multiply so
// EXEC stays all-1s through the WMMA loop (no saveexec regions).
// ---------------------------------------------------------------------------
__global__ __launch_bounds__(32) void collapse_mlp_kernel(
    const float* __restrict__ W1, const float* __restrict__ b1,
    const float* __restrict__ W2, const float* __restrict__ b2,
    float* __restrict__ ws) {
  const int lane  = threadIdx.x;        // 0..31, one full wave, EXEC all-1s
  const int mhalf = lane & 15;
  const bool hi   = lane >= 16;
  const float sel = (mhalf == 0) ? 1.0f : 0.0f;   // B column-0 lane mask

  // b_eff = dot(b1, W2) + b2  (wave reduction)
  float bacc = 0.0f;
  for (int k = lane; k < HIDDEN_C; k += 32) bacc += b1[k] * W2[k];
  #pragma unroll
  for (int off = 16; off >= 1; off >>= 1) bacc += __shfl_xor(bacc, off, 32);
  if (lane == 0) ws[NUM_PAR_C] = bacc + b2[0];

  // w_eff = W1 @ W2 via 5 M-tiles x 40 K-steps of f32 WMMA
  for (int mt = 0; mt < 5; ++mt) {
    const int m0  = mt * 16;
    const int row = m0 + mhalf;                 // W1 row this lane supplies
    v8f acc = {};
    for (int k = 0; k < HIDDEN_C; k += 4) {
      const int ka = k + (hi ? 2 : 0);          // this lane's K pair
      v2f a, b;
      a.x = W1[row * HIDDEN_C + ka];
      a.y = W1[row * HIDDEN_C + ka + 1];
      b.x = W2[ka]     * sel;                   // uniform load + VALU mask
      b.y = W2[ka + 1] * sel;
      acc = __builtin_amdgcn_wmma_f32_16x16x4_f32(
          /*neg_a=*/false, a, /*neg_b=*/false, b,
          /*c_mod=*/(short)0, acc, /*reuse_a=*/false, /*reuse_b=*/false);
    }
    // Column N=0 of D lives in lane 0 (M=m0..m0+7) and lane 16 (M=m0+8..m0+15)
    if (mhalf == 0) {
      const int mb = m0 + (hi ? 8 : 0);
      #pragma unroll
      for (int v = 0; v < 8; ++v) ws[mb + v] = acc[v];
    }
  }
}

// ---------------------------------------------------------------------------
// Kernel 2: per-edge collapsed evaluation (memory-bound path).
// 16 lanes per edge; each lane covers one float4 of the 64-float node rows,
// so each gathered row is a single coalesced 256B segment. x (25.6 MB) is
// L2-resident (192 MB L2); edge_attr (64 MB) / edge_index (16 MB) / out
// (4 MB) stream from HBM -> ~3.6 us floor at 23.3 TB/s.
// ---------------------------------------------------------------------------
__global__ __launch_bounds__(256) void edge_predict_kernel(
    const float* __restrict__ x,
    const long long* __restrict__ edge_index,   // int64, [2, E] flat
    const float* __restrict__ edge_attr,        // [E, 16]
    const float* __restrict__ ws,               // w_eff[80], b_eff at [80]
    float* __restrict__ out) {
  const int tid = blockIdx.x * blockDim.x + threadIdx.x;
  const int e   = tid >> 4;
  const int g   = tid & 15;
  if (e >= N_EDGES_C) return;

  const float4 wn = ((const float4*)ws)[g];     // node weights [4g..4g+3]

  const long long s = edge_index[e];
  const long long d = edge_index[N_EDGES_C + e];
  const float4* xs = (const float4*)(x + s * D_NODE_C);
  const float4* xd = (const float4*)(x + d * D_NODE_C);
  const float4 a = xs[g];
  const float4 b = xd[g];

  float acc = (a.x + b.x) * wn.x + (a.y + b.y) * wn.y +
              (a.z + b.z) * wn.z + (a.w + b.w) * wn.w;

  if (g < 4) {                                  // fold in edge_attr (16 floats)
    const float4 we = ((const float4*)ws)[16 + g];
    const float4 ev = ((const float4*)(edge_attr + (long long)e * D_EDGE_C))[g];
    acc += ev.x * we.x + ev.y * we.y + ev.z * we.z + ev.w * we.w;
  }

  // butterfly reduction across the 16-lane edge group
  #pragma unroll
  for (int off = 8; off >= 1; off >>= 1) acc += __shfl_xor(acc, off, 16);

  if (g == 0) {
    const float z = acc + ws[NUM_PAR_C];
    out[e] = 1.0f / (1.0f + expf(-z));
  }
}

// ---------------------------------------------------------------------------
extern "C" void kernel_launch(void* const* d_in, const int* in_sizes, int n_in,
                              void* d_out, int out_size, void* d_ws, size_t ws_size,
                              hipStream_t stream) {
  const float*     x  = (const float*)d_in[0];
  const long long* ei = (const long long*)d_in[1];
  const float*     ea = (const float*)d_in[2];
  const float*     W1 = (const float*)d_in[3];
  const float*     b1 = (const float*)d_in[4];
  const float*     W2 = (const float*)d_in[5];
  const float*     b2 = (const float*)d_in[6];
  float* ws  = (float*)d_ws;      // 81 floats: w_eff[80] + b_eff
  float* out = (float*)d_out;

  collapse_mlp_kernel<<<dim3(1), dim3(32), 0, stream>>>(W1, b1, W2, b2, ws);

  const long long total = (long long)N_EDGES_C * 16;
  const int blocks = (int)((total + 255) / 256);
  edge_predict_kernel<<<dim3(blocks), dim3(256), 0, stream>>>(x, ei, ea, ws, out);
}